// PrimaryNetwork50_9955734192190
// MI455X (gfx1250) — compile-verified
//
#include <hip/hip_runtime.h>
#include <hip/hip_bf16.h>

// ---------------------------------------------------------------------------
// PrimaryNetwork50 (hypernetwork ResNet-50) forward for MI455X / gfx1250.
// fp32 end-to-end with V_WMMA_F32_16X16X4_F32 (wave32).
// NHWC activations + [z][rs][oc][ic] weights => b64 fragment loads, b128 stores,
// branchless zero-padding (clamped address + 0/1 mask multiply).
// Conv: 4Mx4N register blocking -> 16 WMMA per 8 b64 fragment loads.
// ---------------------------------------------------------------------------

typedef __attribute__((ext_vector_type(2))) float v2f;
typedef __attribute__((ext_vector_type(4))) float v4f;
typedef __attribute__((ext_vector_type(8))) float v8f;
typedef __attribute__((ext_vector_type(4))) unsigned int v4u;
typedef __attribute__((ext_vector_type(8))) int v8i;
typedef __attribute__((ext_vector_type(4))) int v4i;

#define ZTOT   3268
#define ZPAD   3280          // 205 * 16, pads hyper-GEMM1 M dimension
#define WSLICE (9 * 64 * 64) // kt elements per z-index

__device__ __forceinline__ v8f wmma_f32_k4(v2f a, v2f b, v8f c) {
    return __builtin_amdgcn_wmma_f32_16x16x4_f32(false, a, false, b, (short)0, c,
                                                 false, false);
}

// ---------------------------------------------------------------------------
// TDM probe: issue tensor_load_to_lds with a NULL descriptor (D# group0
// count=0 => NULL tensor, hardware NOP) + s_wait_tensorcnt. Runtime no-op,
// exercises the gfx1250 async-tensor issue path.
// ---------------------------------------------------------------------------
#if __has_builtin(__builtin_amdgcn_tensor_load_to_lds)
#define HAVE_TDM 1
__global__ void tdm_probe() {
    __shared__ float lds_pad[64];
    v4u g0 = {};          // count=0 -> NULL tensor descriptor
    v8i g1 = {};          // workgroup_mask=0 (not in cluster)
    v4i g2 = {}, g3 = {};
#if defined(__clang_major__) && (__clang_major__ >= 23)
    __builtin_amdgcn_tensor_load_to_lds(g0, g1, g2, g3, g1, 0);
#else
    __builtin_amdgcn_tensor_load_to_lds(g0, g1, g2, g3, 0);
#endif
    __builtin_amdgcn_s_wait_tensorcnt(0);
    lds_pad[threadIdx.x & 63] = 0.0f;
    __syncthreads();
    (void)lds_pad;
}
#else
#define HAVE_TDM 0
#endif

// ---------------------------------------------------------------------------
// Hypernetwork GEMM 1: h[3280,4096] = z[3268,64] @ hw1[64,4096] + hb1
// ---------------------------------------------------------------------------
__global__ void hyper_gemm1(const float* __restrict__ z,
                            const float* __restrict__ hw1,
                            const float* __restrict__ hb1,
                            float* __restrict__ h) {
    const int lane = threadIdx.x & 31;
    const int half = lane >> 4;
    const int l16  = lane & 15;
    const int m0   = blockIdx.x * 16;
    const int n0   = blockIdx.y * 16;
    const int row  = min(m0 + l16, ZTOT - 1);   // clamped A row (branchless tail)
    const int col  = n0 + l16;
    v8f acc = {};
    for (int kk = 0; kk < 64; kk += 4) {
        const int ka = kk + half * 2;
        v2f a = *(const v2f*)(z + row * 64 + ka);
        v2f b;
        b.x = hw1[ka * 4096 + col];
        b.y = hw1[(ka + 1) * 4096 + col];
        acc = wmma_f32_k4(a, b, acc);
    }
    const float bv = hb1[col];
#pragma unroll
    for (int r = 0; r < 8; ++r)
        h[(size_t)(m0 + r + half * 8) * 4096 + col] = acc[r] + bv;
}

// ---------------------------------------------------------------------------
// Hypernetwork GEMM 2 (batched) fused with weight-layout transform:
//   raw k[n][o][f] = h[n][64,64] @ hw2[64,576] + hb2,   f = ic*9 + rs
//   stored as kt[((n*9 + rs)*64 + o)*64 + ic]
// ---------------------------------------------------------------------------
__global__ void hyper_gemm2(const float* __restrict__ h,
                            const float* __restrict__ hw2,
                            const float* __restrict__ hb2,
                            float* __restrict__ kt) {
    const int n    = blockIdx.z;
    const int lane = threadIdx.x & 31;
    const int half = lane >> 4;
    const int l16  = lane & 15;
    const int m0   = blockIdx.x * 16;
    const int col  = blockIdx.y * 16 + l16;     // f in [0,576)
    const float* A = h + (size_t)n * 4096;
    v8f acc = {};
    for (int kk = 0; kk < 64; kk += 4) {
        const int ka = kk + half * 2;
        v2f a = *(const v2f*)(A + (m0 + l16) * 64 + ka);
        v2f b;
        b.x = hw2[ka * 576 + col];
        b.y = hw2[(ka + 1) * 576 + col];
        acc = wmma_f32_k4(a, b, acc);
    }
    const float bv = hb2[col];
    const int ic = col / 9;
    const int rs = col - ic * 9;
    float* dst = kt + ((size_t)n * 9 + rs) * 4096 + ic;
#pragma unroll
    for (int r = 0; r < 8; ++r) {
        const int o = m0 + r + half * 8;
        dst[o * 64] = acc[r] + bv;
    }
}

// ---------------------------------------------------------------------------
// Implicit-GEMM 3x3 conv (pad 1), NHWC in/out, transformed weights.
// One wave computes 64 (Cout) x 64 (pixels): 4 M-tiles x 4 N-tiles,
// 16 accumulators; per K-step 4 A-b64 + 4 B-b64 loads feed 16 WMMAs.
//   W(oc,ic,rs) = kt[((zoff + (oc>>6)*kgrp + (ic>>6))*9 + rs)*4096
//                    + (oc&63)*64 + (ic&63)]
// ---------------------------------------------------------------------------
__global__ void conv3x3_wmma(const float* __restrict__ in, float* __restrict__ out,
                             const float* __restrict__ kt, int zoff, int kgrp,
                             int Cin, int Cout, int Hin, int Ho, int stride) {
    const int lane = threadIdx.x & 31;
    const int half = lane >> 4;
    const int l16  = lane & 15;
    const int m0   = blockIdx.x * 64;
    const int cb   = blockIdx.y * 64;
    const int Win  = Hin, Wo = Ho, HWo = Ho * Wo;

    // per-lane column state for the 4 N-tiles
    int nimg[4], oy[4], ox[4], cbase[4];
#pragma unroll
    for (int j = 0; j < 4; ++j) {
        const int col = cb + j * 16 + l16;
        nimg[j] = col / HWo;
        const int rem = col - nimg[j] * HWo;
        oy[j] = rem / Wo;
        ox[j] = rem - oy[j] * Wo;
        cbase[j] = col;                 // flat pixel index == NHWC pixel offset
    }

    // weight row bases for the 4 M-tiles (oc = A row this lane loads)
    const float* wbase[4];
#pragma unroll
    for (int m = 0; m < 4; ++m) {
        const int oc = m0 + m * 16 + l16;
        wbase[m] = kt + ((size_t)(zoff + (oc >> 6) * kgrp) * 9) * 4096 +
                   (oc & 63) * 64;
    }

    v8f acc[4][4];
#pragma unroll
    for (int m = 0; m < 4; ++m)
#pragma unroll
        for (int j = 0; j < 4; ++j) acc[m][j] = (v8f){};

    const int half2 = half * 2;
    for (int rs = 0; rs < 9; ++rs) {
        const int r = rs / 3, s = rs - (rs / 3) * 3;
        float msk[4];
        int   base[4];
#pragma unroll
        for (int j = 0; j < 4; ++j) {
            const int iy = oy[j] * stride + r - 1;
            const int ix = ox[j] * stride + s - 1;
            const bool ok = (iy >= 0) && (iy < Hin) && (ix >= 0) && (ix < Win);
            const int iyc = min(max(iy, 0), Hin - 1);
            const int ixc = min(max(ix, 0), Win - 1);
            base[j] = ((nimg[j] * Hin + iyc) * Win + ixc) * Cin;   // NHWC
            msk[j]  = ok ? 1.0f : 0.0f;
        }
        const float* wrs[4];
#pragma unroll
        for (int m = 0; m < 4; ++m) wrs[m] = wbase[m] + rs * 4096;
        // speculative prefetch of the next tap's weight lines
        __builtin_prefetch(wrs[0] + 4096, 0, 1);
        __builtin_prefetch(wrs[3] + 4096, 0, 1);

        for (int ic0 = 0; ic0 < Cin; ic0 += 4) {
            const int ka = ic0 + half2;
            const int g  = (ka >> 6) * WSLICE;   // crosses 64-ic groups
            const int kl = ka & 63;
            v2f a[4];
#pragma unroll
            for (int m = 0; m < 4; ++m) a[m] = *(const v2f*)(wrs[m] + g + kl);
#pragma unroll
            for (int j = 0; j < 4; ++j) {
                v2f b = *(const v2f*)(in + base[j] + ka);  // clamped => in-bounds
                b *= msk[j];                                // branchless zero-pad
#pragma unroll
                for (int m = 0; m < 4; ++m)
                    acc[m][j] = wmma_f32_k4(a[m], b, acc[m][j]);
            }
        }
    }

    // store: lane owns pixel cbase[j]; 8 consecutive channels per acc => 2x b128
#pragma unroll
    for (int m = 0; m < 4; ++m)
#pragma unroll
        for (int j = 0; j < 4; ++j) {
            float* p = out + (size_t)cbase[j] * Cout + m0 + m * 16 + half * 8;
            v4f lo = {acc[m][j][0], acc[m][j][1], acc[m][j][2], acc[m][j][3]};
            v4f hi = {acc[m][j][4], acc[m][j][5], acc[m][j][6], acc[m][j][7]};
            *(v4f*)p       = lo;
            *(v4f*)(p + 4) = hi;
        }
}

// ---------------------------------------------------------------------------
// BatchNorm stats, NHWC: per-channel sum/sumsq. grid (C/64 [min 1], chunks).
// 256 threads = 64 channels x 4 spatial partials; LDS reduce; 1 atomic pair.
// ---------------------------------------------------------------------------
__global__ void bn_stats_nhwc(const float* __restrict__ x, float* __restrict__ stats,
                              int C, int NHW) {
    const int c = blockIdx.x * 64 + (threadIdx.x & 63);
    const int t = threadIdx.x >> 6;  // 0..3
    float s = 0.0f, q = 0.0f;
    if (c < C) {
        for (int i = blockIdx.y * 4 + t; i < NHW; i += gridDim.y * 4) {
            const float v = x[(size_t)i * C + c];
            s += v;
            q += v * v;
        }
    }
    __shared__ float ss[256], sq[256];
    ss[threadIdx.x] = s;
    sq[threadIdx.x] = q;
    __syncthreads();
    if (t == 0 && c < C) {
        const int u = threadIdx.x;
        s = ss[u] + ss[u + 64] + ss[u + 128] + ss[u + 192];
        q = sq[u] + sq[u + 64] + sq[u + 128] + sq[u + 192];
        atomicAdd(&stats[2 * c], s);
        atomicAdd(&stats[2 * c + 1], q);
    }
}

// y = (x - m) * rsqrt(v + 1e-5) [+res] [relu];  NHWC, C = power of two
__global__ void bn_apply_nhwc(const float* __restrict__ x,
                              const float* __restrict__ stats,
                              const float* __restrict__ res, float* __restrict__ out,
                              int Cmask, float invCnt, int relu, int total) {
    const int i = blockIdx.x * blockDim.x + threadIdx.x;
    if (i >= total) return;
    const int c = i & Cmask;
    const float m = stats[2 * c] * invCnt;
    const float v = stats[2 * c + 1] * invCnt - m * m;
    float y = (x[i] - m) * rsqrtf(v + 1e-5f);
    if (res) y += res[i];
    if (relu) y = fmaxf(y, 0.0f);
    out[i] = y;
}

// ---------------------------------------------------------------------------
// data_bn: stats over NCHW input x [64,3,32,32] (C=3)
// ---------------------------------------------------------------------------
__global__ void bn_stats_nchw(const float* __restrict__ x, float* __restrict__ stats,
                              int C, int HW) {
    const int c = blockIdx.x;
    __shared__ float ssum[256], ssq[256];
    float s = 0.0f, q = 0.0f;
    const int total = 64 * HW;
    for (int i = blockIdx.y * blockDim.x + threadIdx.x; i < total;
         i += gridDim.y * blockDim.x) {
        const int n = i / HW;
        const int j = i - n * HW;
        const float v = x[((size_t)n * C + c) * HW + j];
        s += v;
        q += v * v;
    }
    ssum[threadIdx.x] = s;
    ssq[threadIdx.x]  = q;
    __syncthreads();
    for (int off = 128; off > 0; off >>= 1) {
        if ((int)threadIdx.x < off) {
            ssum[threadIdx.x] += ssum[threadIdx.x + off];
            ssq[threadIdx.x]  += ssq[threadIdx.x + off];
        }
        __syncthreads();
    }
    if (threadIdx.x == 0) {
        atomicAdd(&stats[2 * c],     ssum[0]);
        atomicAdd(&stats[2 * c + 1], ssq[0]);
    }
}

// data_bn apply: NCHW x [64,3,32,32] -> NHWC xbn [64,32,32,3]
__global__ void data_bn_apply(const float* __restrict__ x,
                              const float* __restrict__ stats,
                              float* __restrict__ xbn) {
    const int idx = blockIdx.x * blockDim.x + threadIdx.x;  // 196608
    if (idx >= 64 * 3 * 32 * 32) return;
    const int xi = idx & 31;
    const int y  = (idx >> 5) & 31;
    const int c  = (idx >> 10) % 3;
    const int n  = idx / 3072;
    const float invCnt = 1.0f / (64.0f * 1024.0f);
    const float m = stats[2 * c] * invCnt;
    const float v = stats[2 * c + 1] * invCnt - m * m;
    xbn[((n * 32 + y) * 32 + xi) * 3 + c] = (x[idx] - m) * rsqrtf(v + 1e-5f);
}

// Stem conv: NHWC xbn [64,32,32,3] -> NHWC raw [64,32,32,64]
__global__ void stem_conv(const float* __restrict__ xbn, const float* __restrict__ w,
                          float* __restrict__ out) {
    const int idx = blockIdx.x * blockDim.x + threadIdx.x;  // 64*32*32*64
    if (idx >= 64 * 32 * 32 * 64) return;
    const int oc = idx & 63;
    const int x  = (idx >> 6) & 31;
    const int y  = (idx >> 11) & 31;
    const int n  = idx >> 16;
    float s = 0.0f;
    for (int r = 0; r < 3; ++r)
        for (int t = 0; t < 3; ++t) {
            const int iy = y + r - 1, ix = x + t - 1;
            if (iy >= 0 && iy < 32 && ix >= 0 && ix < 32) {
                const float* px = xbn + ((n * 32 + iy) * 32 + ix) * 3;
                for (int c = 0; c < 3; ++c)
                    s += px[c] * w[((oc * 3 + c) * 3 + r) * 3 + t];
            }
        }
    out[idx] = s;
}

// avgpool NHWC [64,4,4,2048] -> pooled [64,2048]
__global__ void avgpool(const float* __restrict__ x, float* __restrict__ pooled,
                        int HW) {
    const int i = blockIdx.x * blockDim.x + threadIdx.x;  // 64*2048
    if (i >= 64 * 2048) return;
    const int n = i >> 11;
    const int c = i & 2047;
    float s = 0.0f;
    for (int j = 0; j < HW; ++j) s += x[((size_t)n * HW + j) * 2048 + c];
    pooled[i] = s * (1.0f / HW);
}

// out[64,10] = pooled[64,2048] @ final_w[10,2048]^T + final_b
__global__ void fc_kernel(const float* __restrict__ pooled, const float* __restrict__ w,
                          const float* __restrict__ b, float* __restrict__ out) {
    const int i = blockIdx.x * blockDim.x + threadIdx.x;  // 640
    if (i >= 640) return;
    const int n = i / 10, j = i - (i / 10) * 10;
    float s = b[j];
    for (int k = 0; k < 2048; ++k) s += pooled[n * 2048 + k] * w[j * 2048 + k];
    out[i] = s;
}

// ---------------------------------------------------------------------------
// Host orchestration
// ---------------------------------------------------------------------------
static const int ZS[52][2] = {
    {1,1},{1,1},{4,1},{4,1},{1,4},{1,1},{4,1},{1,4},{1,1},{4,1},
    {2,4},{2,2},{8,2},{8,4},{2,8},{2,2},{8,2},{2,8},{2,2},{8,2},
    {2,8},{2,2},{8,2},{4,8},{4,4},{16,4},{16,8},{4,16},{4,4},{16,4},
    {4,16},{4,4},{16,4},{4,16},{4,4},{16,4},{4,16},{4,4},{16,4},{4,16},
    {4,4},{16,4},{8,16},{8,8},{32,8},{32,16},{8,32},{8,8},{32,8},{8,32},
    {8,8},{32,8}
};

static void run_bn(const float* raw, float* dst, const float* res, int C, int HW,
                   int relu, float* stats, hipStream_t stream) {
    hipMemsetAsync(stats, 0, 2 * C * sizeof(float), stream);
    dim3 g((C + 63) / 64, 8);
    bn_stats_nhwc<<<g, 256, 0, stream>>>(raw, stats, C, 64 * HW);
    const int total = 64 * C * HW;
    const float invCnt = 1.0f / (64.0f * HW);
    bn_apply_nhwc<<<(total + 255) / 256, 256, 0, stream>>>(raw, stats, res, dst,
                                                           C - 1, invCnt, relu, total);
}

static void run_conv(const float* in, float* out, const float* kt, int zoff,
                     int kgrp, int Cin, int Cout, int Hin, int Ho, int stride,
                     hipStream_t stream) {
    dim3 grid(Cout / 64, (64 * Ho * Ho) / 64);
    conv3x3_wmma<<<grid, 32, 0, stream>>>(in, out, kt, zoff, kgrp, Cin, Cout,
                                          Hin, Ho, stride);
}

extern "C" void kernel_launch(void* const* d_in, const int* in_sizes, int n_in,
                              void* d_out, int out_size, void* d_ws, size_t ws_size,
                              hipStream_t stream) {
    const float* x       = (const float*)d_in[0];
    const float* conv1_w = (const float*)d_in[1];
    const float* z_all   = (const float*)d_in[2];
    const float* hw1     = (const float*)d_in[3];
    const float* hb1     = (const float*)d_in[4];
    const float* hw2     = (const float*)d_in[5];
    const float* hb2     = (const float*)d_in[6];
    const float* final_w = (const float*)d_in[7];
    const float* final_b = (const float*)d_in[8];
    float* out = (float*)d_out;

    // workspace layout
    char* ws = (char*)d_ws;
    size_t off = 0;
    float* kt   = (float*)(ws + off); off += (size_t)ZTOT * WSLICE * 4;      // 482 MB
    float* hbuf = (float*)(ws + off); off += (size_t)ZPAD * 4096 * 4;        // 53.7 MB
    float* act0 = (float*)(ws + off); off += (size_t)64 * 32 * 32 * 256 * 4; // 64 MB
    float* act1 = (float*)(ws + off); off += (size_t)64 * 32 * 32 * 256 * 4;
    float* act2 = (float*)(ws + off); off += (size_t)64 * 32 * 32 * 256 * 4;
    float* act3 = (float*)(ws + off); off += (size_t)64 * 32 * 32 * 256 * 4;
    float* xbn  = (float*)(ws + off); off += (size_t)64 * 32 * 32 * 3 * 4;
    float* stats  = (float*)(ws + off); off += 2 * 2048 * 4;
    float* pooled = (float*)(ws + off); off += (size_t)64 * 2048 * 4;

    int zoff[52];
    {
        int acc = 0;
        for (int e = 0; e < 52; ++e) { zoff[e] = acc; acc += ZS[e][0] * ZS[e][1]; }
    }

#if HAVE_TDM
    tdm_probe<<<1, 64, 0, stream>>>();
#endif

    // 1) hypernetwork weight generation (GEMM2 fused with layout transform)
    {
        dim3 g1(ZPAD / 16, 4096 / 16);
        hyper_gemm1<<<g1, 32, 0, stream>>>(z_all, hw1, hb1, hbuf);
        dim3 g2(64 / 16, 576 / 16, ZTOT);
        hyper_gemm2<<<g2, 32, 0, stream>>>(hbuf, hw2, hb2, kt);
    }

    // 2) data_bn (NCHW->NHWC) + stem conv + bn/relu
    hipMemsetAsync(stats, 0, 2 * 3 * sizeof(float), stream);
    {
        dim3 g(3, 8);
        bn_stats_nchw<<<g, 256, 0, stream>>>(x, stats, 3, 32 * 32);
    }
    data_bn_apply<<<(64 * 3 * 32 * 32 + 255) / 256, 256, 0, stream>>>(x, stats, xbn);
    stem_conv<<<(64 * 32 * 32 * 64) / 256, 256, 0, stream>>>(xbn, conv1_w, act1);
    run_bn(act1, act0, nullptr, 64, 32 * 32, 1, stats, stream);

    // 3) 16 bottleneck blocks (all tensors NHWC)
    float* cur = act0;
    float* nxt = act3;
    float* t1  = act1;
    float* t2  = act2;
    int cin = 64, hin = 32;
    int j = 0;
    for (int i = 0; i < 16; ++i) {
        const int e1 = j, e2 = j + 1, e3 = j + 2;
        int es = -1, stride = 1;
        if (i == 0)                           { es = j + 3; stride = 1; j += 4; }
        else if (i == 3 || i == 7 || i == 13) { es = j + 3; stride = 2; j += 4; }
        else                                  { j += 3; }
        const int c1 = ZS[e1][0] * 64;
        const int c3 = ZS[e3][0] * 64;
        const int hout = hin / stride;

        run_conv(cur, t1, kt, zoff[e1], ZS[e1][1], cin, c1, hin, hin, 1, stream);
        run_bn(t1, t1, nullptr, c1, hin * hin, 1, stats, stream);

        run_conv(t1, t2, kt, zoff[e2], ZS[e2][1], c1, c1, hin, hout, stride, stream);
        run_bn(t2, t2, nullptr, c1, hout * hout, 1, stats, stream);

        run_conv(t2, t1, kt, zoff[e3], ZS[e3][1], c1, c3, hout, hout, 1, stream);

        const float* sc = cur;
        if (es >= 0) {
            run_conv(cur, t2, kt, zoff[es], ZS[es][1], cin, c3, hin, hout, stride,
                     stream);
            run_bn(t2, t2, nullptr, c3, hout * hout, 0, stats, stream);
            sc = t2;
        }
        run_bn(t1, nxt, sc, c3, hout * hout, 1, stats, stream);

        float* tmp = cur; cur = nxt; nxt = tmp;
        cin = c3; hin = hout;
    }

    // 4) avgpool + fc  (cur: NHWC [64,4,4,2048])
    avgpool<<<(64 * 2048) / 256, 256, 0, stream>>>(cur, pooled, hin * hin);
    fc_kernel<<<10, 64, 0, stream>>>(pooled, final_w, final_b, out);

    (void)in_sizes; (void)n_in; (void)out_size; (void)ws_size; (void)off;
}